// VolumeRenderer_43722767073725
// MI455X (gfx1250) — compile-verified
//
#include <hip/hip_runtime.h>
#include <stdint.h>

// -------- problem constants (from reference) --------
#define N_RAYS 131072
#define N_PTS  128
#define FARV   1e10f

// -------- launch geometry --------
#define WAVES_PER_BLOCK 8
#define RAYS_PER_WAVE   8
#define THREADS        (WAVES_PER_BLOCK * 32)

// per-ray staging buffer in LDS: density 512B + depth 512B + feature 1536B
#define DENS_OFF   0
#define DEPTH_OFF  512
#define FEAT_OFF   1024
#define BUF_BYTES  2560
#define WAVE_LDS   (2 * BUF_BYTES)                 // double buffer
#define SMEM_BYTES (WAVES_PER_BLOCK * WAVE_LDS)    // 40960 B / workgroup

// ---- CDNA5 async global->LDS copy (ASYNCcnt-tracked), GV addressing ----
__device__ __forceinline__ void async_ld_b128(uint32_t lds_off, const void* gptr) {
    asm volatile("global_load_async_to_lds_b128 %0, %1, off"
                 :: "v"(lds_off), "v"((uint64_t)(uintptr_t)gptr)
                 : "memory");
}

// Stage one ray (2.5 KB) into LDS with 5 fully-coalesced async B128 ops/wave.
__device__ __forceinline__ void stage_ray(uint32_t buf_base, int lane,
                                          const float* __restrict__ density,
                                          const float* __restrict__ depthv,
                                          const float* __restrict__ feature,
                                          int ray) {
    const char* dsrc = (const char*)(density + (size_t)ray * N_PTS) + lane * 16;
    const char* zsrc = (const char*)(depthv  + (size_t)ray * N_PTS) + lane * 16;
    const char* fsrc = (const char*)(feature + (size_t)ray * N_PTS * 3);
    async_ld_b128(buf_base + DENS_OFF  + lane * 16, dsrc);
    async_ld_b128(buf_base + DEPTH_OFF + lane * 16, zsrc);
    // 1536 B of per-sample RGB as three coalesced 512 B sweeps
    async_ld_b128(buf_base + FEAT_OFF        + lane * 16, fsrc        + lane * 16);
    async_ld_b128(buf_base + FEAT_OFF + 512  + lane * 16, fsrc + 512  + lane * 16);
    async_ld_b128(buf_base + FEAT_OFF + 1024 + lane * 16, fsrc + 1024 + lane * 16);
}

// Process one staged ray: per-lane 4 samples, wave32 exclusive prefix-product
// scan for transmittance, butterfly reductions, lane 0 writes results.
__device__ __forceinline__ void compute_ray(const unsigned char* buf, int lane,
                                            int ray, float* __restrict__ out) {
    const float4 d4 = *(const float4*)(buf + DENS_OFF  + lane * 16);
    const float4 z4 = *(const float4*)(buf + DEPTH_OFF + lane * 16);
    const float4 f0 = *(const float4*)(buf + FEAT_OFF + lane * 48);
    const float4 f1 = *(const float4*)(buf + FEAT_OFF + lane * 48 + 16);
    const float4 f2 = *(const float4*)(buf + FEAT_OFF + lane * 48 + 32);

    // deltas (next lane's first depth via shuffle; last sample gets FAR)
    float znext = __shfl_down(z4.x, 1, 32);
    float del0 = z4.y - z4.x;
    float del1 = z4.z - z4.y;
    float del2 = z4.w - z4.z;
    float del3 = (lane == 31) ? FARV : (znext - z4.w);

    float t0 = __expf(-d4.x * del0);
    float t1 = __expf(-d4.y * del1);
    float t2 = __expf(-d4.z * del2);
    float t3 = __expf(-d4.w * del3);

    // intra-lane exclusive prefix products
    float e1 = t0, e2 = t0 * t1, e3 = e2 * t2;
    float L  = e3 * t3;                       // lane-local total product

    // wave32 inclusive scan (product) -> exclusive
    float p = L;
#pragma unroll
    for (int d = 1; d < 32; d <<= 1) {
        float n = __shfl_up(p, d, 32);
        if (lane >= d) p *= n;
    }
    float excl = __shfl_up(p, 1, 32);
    if (lane == 0) excl = 1.0f;

    float T0 = excl, T1 = excl * e1, T2 = excl * e2, T3 = excl * e3;
    float w0 = T0 * (1.0f - t0), w1 = T1 * (1.0f - t1);
    float w2 = T2 * (1.0f - t2), w3 = T3 * (1.0f - t3);

    // RGB accumulation (sample s rgb laid out at floats 3s..3s+2)
    float r = w0 * f0.x + w1 * f0.w + w2 * f1.z + w3 * f2.y;
    float g = w0 * f0.y + w1 * f1.x + w2 * f1.w + w3 * f2.z;
    float b = w0 * f0.z + w1 * f1.y + w2 * f2.x + w3 * f2.w;

    // depth estimate: select samples with T < 1
    float St = 0.0f, Std = 0.0f;
    if (T0 < 1.0f) { St += t0; Std += t0 * z4.x; }
    if (T1 < 1.0f) { St += t1; Std += t1 * z4.y; }
    if (T2 < 1.0f) { St += t2; Std += t2 * z4.z; }
    if (T3 < 1.0f) { St += t3; Std += t3 * z4.w; }

    // wave butterfly reduction of 5 accumulators
#pragma unroll
    for (int off = 16; off > 0; off >>= 1) {
        r   += __shfl_xor(r,   off, 32);
        g   += __shfl_xor(g,   off, 32);
        b   += __shfl_xor(b,   off, 32);
        St  += __shfl_xor(St,  off, 32);
        Std += __shfl_xor(Std, off, 32);
    }

    if (lane == 0) {
        out[3 * ray + 0] = r;
        out[3 * ray + 1] = g;
        out[3 * ray + 2] = b;
        float dep = Std / St;                 // 0/0 -> NaN, matches jnp path
        out[3 * N_RAYS + ray] = (dep == 0.0f) ? __builtin_inff() : dep;
    }
}

__global__ __launch_bounds__(THREADS)
void VolumeRenderer_43722767073725_kernel(const float* __restrict__ density,
                                          const float* __restrict__ feature,
                                          const float* __restrict__ depthv,
                                          float* __restrict__ out) {
    __shared__ __align__(16) unsigned char smem[SMEM_BYTES];

    const int lane = threadIdx.x & 31;
    const int wave = threadIdx.x >> 5;
    unsigned char* wave_buf = smem + wave * WAVE_LDS;
    // generic-pointer low 32 bits == LDS byte offset (addrspace(3) origin 0)
    const uint32_t wave_lds = (uint32_t)(uintptr_t)wave_buf;

    const int ray0 = (blockIdx.x * WAVES_PER_BLOCK + wave) * RAYS_PER_WAVE;

    // software pipeline: prefetch ray0, then overlap stage(i+1) with compute(i)
    stage_ray(wave_lds, lane, density, depthv, feature, ray0);

#pragma unroll
    for (int i = 0; i < RAYS_PER_WAVE; ++i) {
        if (i + 1 < RAYS_PER_WAVE) {
            stage_ray(wave_lds + ((i + 1) & 1) * BUF_BYTES, lane,
                      density, depthv, feature, ray0 + i + 1);
            // 5 newest async ops (next ray) may stay in flight; ray i is done
            asm volatile("s_wait_asynccnt 5" ::: "memory");
        } else {
            asm volatile("s_wait_asynccnt 0" ::: "memory");
        }
        compute_ray(wave_buf + (i & 1) * BUF_BYTES, lane, ray0 + i, out);
    }
}

extern "C" void kernel_launch(void* const* d_in, const int* in_sizes, int n_in,
                              void* d_out, int out_size, void* d_ws, size_t ws_size,
                              hipStream_t stream) {
    (void)in_sizes; (void)n_in; (void)out_size; (void)d_ws; (void)ws_size;
    const float* density = (const float*)d_in[0];   // [N_RAYS, N_PTS]
    const float* feature = (const float*)d_in[1];   // [N_RAYS, N_PTS, 3]
    const float* depthv  = (const float*)d_in[2];   // [N_RAYS, N_PTS]
    float* out = (float*)d_out;                     // [N_RAYS*3] feat ++ [N_RAYS] depth

    const int blocks = N_RAYS / (WAVES_PER_BLOCK * RAYS_PER_WAVE); // 2048
    VolumeRenderer_43722767073725_kernel<<<blocks, THREADS, 0, stream>>>(
        density, feature, depthv, out);
}